// GCN_27204322853153
// MI455X (gfx1250) — compile-verified
//
#include <hip/hip_runtime.h>
#include <hip/hip_bf16.h>

#define N_NODES   50000
#define N_EDGES   800000
#define N_GRAPHS  128
#define EMBED     256
#define OUTD      128
#define LAYERS    5
#define ATOM_COLS 9
#define ATOM_VOCAB 128
#define BOND_COLS 3
#define BOND_VOCAB 8
#define BN_EPS    1e-5f

typedef __attribute__((ext_vector_type(2))) float v2f;
typedef __attribute__((ext_vector_type(8))) float v8f;

// ---------------------------------------------------------------- utilities
__global__ void zero_f32(float* __restrict__ p, int n) {
  int i = blockIdx.x * blockDim.x + threadIdx.x;
  int stride = gridDim.x * blockDim.x;
  for (; i < n; i += stride) p[i] = 0.0f;
}

// ------------------------------------------------------------- atom encoder
// h[n][d] = sum_c atom_emb[c][nfeat[n][c]][d]; one thread per (node, float4)
__global__ void atom_encoder(const int* __restrict__ nfeat,
                             const float* __restrict__ atom_emb,
                             float* __restrict__ h) {
  int gid = blockIdx.x * blockDim.x + threadIdx.x;
  if (gid >= N_NODES * 64) return;
  int node = gid >> 6;
  int d = (gid & 63) << 2;
  float4 acc = make_float4(0.f, 0.f, 0.f, 0.f);
#pragma unroll
  for (int c = 0; c < ATOM_COLS; ++c) {
    int f = nfeat[node * ATOM_COLS + c];
    const float4 v =
        *(const float4*)(atom_emb + ((size_t)(c * ATOM_VOCAB + f)) * EMBED + d);
    acc.x += v.x; acc.y += v.y; acc.z += v.z; acc.w += v.w;
  }
  *(float4*)(h + (size_t)node * EMBED + d) = acc;
}

// ------------------------------------------------------------------ degrees
__global__ void degs_init(float* __restrict__ degs) {
  int i = blockIdx.x * blockDim.x + threadIdx.x;
  if (i < N_NODES) degs[i] = 1.0f;
}
__global__ void degs_acc(const int* __restrict__ dst, float* __restrict__ degs) {
  int e = blockIdx.x * blockDim.x + threadIdx.x;
  if (e < N_EDGES) atomicAdd(&degs[dst[e]], 1.0f);
}

// -------------------------------------------------------- edge message pass
// neigh[dst[e]] += h[src[e]] + sum_c edge_emb[l][c][efeat[e][c]]
// one thread per (edge, float4 chunk); gathers hit L2 (h fits in 192MB L2)
__global__ void edge_message(const float* __restrict__ h,
                             const int* __restrict__ efeat,
                             const int* __restrict__ src,
                             const int* __restrict__ dst,
                             const float* __restrict__ eemb,  // [3][8][256]
                             float* __restrict__ neigh) {
  int gid = blockIdx.x * blockDim.x + threadIdx.x;
  if (gid >= N_EDGES * 64) return;
  int e = gid >> 6;
  int d = (gid & 63) << 2;
  int s = src[e];
  int t = dst[e];
  float4 acc = *(const float4*)(h + (size_t)s * EMBED + d);
#pragma unroll
  for (int c = 0; c < BOND_COLS; ++c) {
    int f = efeat[e * BOND_COLS + c];
    const float4 v =
        *(const float4*)(eemb + (size_t)(c * BOND_VOCAB + f) * EMBED + d);
    acc.x += v.x; acc.y += v.y; acc.z += v.z; acc.w += v.w;
  }
  float* out = neigh + (size_t)t * EMBED + d;
  atomicAdd(out + 0, acc.x);
  atomicAdd(out + 1, acc.y);
  atomicAdd(out + 2, acc.z);
  atomicAdd(out + 3, acc.w);
}

// -------------------------------------------------------- layer GEMM (WMMA)
// hnew = ((h + neigh)/degs) @ W + b, written in place into `neigh`.
// Per-column sum / sum-of-squares accumulated for batch norm.
// Block = 256 thr (8 waves); block owns a 16-row stripe; stages T into LDS;
// each wave computes two 16x16 output tiles with V_WMMA_F32_16X16X4_F32.
__global__ void gemm_layer(const float* __restrict__ h,
                           float* __restrict__ neigh,  // in: neigh / out: hnew
                           const float* __restrict__ degs,
                           const float* __restrict__ W,     // [256][256] k-major
                           const float* __restrict__ bias,  // [256]
                           float* __restrict__ colsum,
                           float* __restrict__ colsumsq) {
  __shared__ float T[16][EMBED];  // 16 KB
  const int row0 = blockIdx.x * 16;
  const int tid = threadIdx.x;

  // stage T = (h + neigh) / degs for this 16-row stripe
  for (int i = tid; i < 16 * EMBED; i += 256) {
    int r = i >> 8;
    int c = i & (EMBED - 1);
    size_t idx = (size_t)(row0 + r) * EMBED + c;
    T[r][c] = (h[idx] + neigh[idx]) / degs[row0 + r];
  }
  __syncthreads();

  const int wave = tid >> 5;
  const int lane = tid & 31;
  const int m = lane & 15;               // A row
  const int kb = (lane >> 4) << 1;       // K sub-offset (0 or 2)
  const int n = lane & 15;               // B/C column
  const int mbase = (lane >> 4) << 3;    // C row base (0 or 8)

  for (int t = 0; t < 2; ++t) {
    const int ncol = (wave * 2 + t) * 16;
    v8f acc = {};
    for (int k0 = 0; k0 < EMBED; k0 += 4) {
      v2f afrag, bfrag;
      afrag.x = T[m][k0 + kb];
      afrag.y = T[m][k0 + kb + 1];
      bfrag.x = W[(size_t)(k0 + kb) * EMBED + ncol + n];
      bfrag.y = W[(size_t)(k0 + kb + 1) * EMBED + ncol + n];
      acc = __builtin_amdgcn_wmma_f32_16x16x4_f32(
          false, afrag, false, bfrag, (short)0, acc, false, false);
    }
    const float bv = bias[ncol + n];
    float s = 0.f, sq = 0.f;
#pragma unroll
    for (int r = 0; r < 8; ++r) {
      float v = acc[r] + bv;
      neigh[(size_t)(row0 + mbase + r) * EMBED + ncol + n] = v;
      s += v;
      sq += v * v;
    }
    atomicAdd(&colsum[ncol + n], s);
    atomicAdd(&colsumsq[ncol + n], sq);
  }
}

// ----------------------------------------------------------- batchnorm prep
__global__ void bn_params(const float* __restrict__ colsum,
                          const float* __restrict__ colsumsq,
                          const float* __restrict__ gamma,
                          const float* __restrict__ beta,
                          float* __restrict__ scale,
                          float* __restrict__ shift) {
  int c = threadIdx.x;  // 256 threads
  const float inv_n = 1.0f / (float)N_NODES;
  float mu = colsum[c] * inv_n;
  float var = colsumsq[c] * inv_n - mu * mu;
  var = fmaxf(var, 0.0f);
  float sc = gamma[c] * rsqrtf(var + BN_EPS);
  scale[c] = sc;
  shift[c] = beta[c] - mu * sc;
}

// ---------------------------------------------------------- batchnorm+ReLU
__global__ void bn_relu(const float* __restrict__ hnew,
                        const float* __restrict__ scale,
                        const float* __restrict__ shift,
                        float* __restrict__ h) {
  int i = blockIdx.x * blockDim.x + threadIdx.x;
  if (i >= N_NODES * EMBED) return;
  int c = i & (EMBED - 1);
  h[i] = fmaxf(0.0f, hnew[i] * scale[c] + shift[c]);
}

// -------------------------------------------------------------- graph pool
__global__ void pool_sum(const float* __restrict__ h,
                         const int* __restrict__ gids,
                         float* __restrict__ gsum) {
  int gid = blockIdx.x * blockDim.x + threadIdx.x;
  if (gid >= N_NODES * 64) return;
  int node = gid >> 6;
  int d = (gid & 63) << 2;
  int g = gids[node];
  const float4 v = *(const float4*)(h + (size_t)node * EMBED + d);
  float* out = gsum + (size_t)g * EMBED + d;
  atomicAdd(out + 0, v.x);
  atomicAdd(out + 1, v.y);
  atomicAdd(out + 2, v.z);
  atomicAdd(out + 3, v.w);
}
__global__ void pool_cnt(const int* __restrict__ gids, float* __restrict__ gcnt) {
  int nn = blockIdx.x * blockDim.x + threadIdx.x;
  if (nn < N_NODES) atomicAdd(&gcnt[gids[nn]], 1.0f);
}

// ------------------------------------------------- final projection (WMMA)
// out[128][128] = (gsum/max(cnt,1)) @ Wp + bp ; 8 blocks x 16-row stripes,
// 8 waves, one 16x16 tile each (OUTD = 128).
__global__ void final_gemm(const float* __restrict__ gsum,
                           const float* __restrict__ gcnt,
                           const float* __restrict__ Wp,  // [256][128]
                           const float* __restrict__ bp,  // [128]
                           float* __restrict__ out) {
  __shared__ float G[16][EMBED];
  const int row0 = blockIdx.x * 16;
  const int tid = threadIdx.x;
  for (int i = tid; i < 16 * EMBED; i += 256) {
    int r = i >> 8;
    int c = i & (EMBED - 1);
    float cnt = fmaxf(gcnt[row0 + r], 1.0f);
    G[r][c] = gsum[(size_t)(row0 + r) * EMBED + c] / cnt;
  }
  __syncthreads();

  const int wave = tid >> 5;
  const int lane = tid & 31;
  const int m = lane & 15;
  const int kb = (lane >> 4) << 1;
  const int n = lane & 15;
  const int mbase = (lane >> 4) << 3;
  const int ncol = wave * 16;

  v8f acc = {};
  for (int k0 = 0; k0 < EMBED; k0 += 4) {
    v2f afrag, bfrag;
    afrag.x = G[m][k0 + kb];
    afrag.y = G[m][k0 + kb + 1];
    bfrag.x = Wp[(size_t)(k0 + kb) * OUTD + ncol + n];
    bfrag.y = Wp[(size_t)(k0 + kb + 1) * OUTD + ncol + n];
    acc = __builtin_amdgcn_wmma_f32_16x16x4_f32(
        false, afrag, false, bfrag, (short)0, acc, false, false);
  }
  const float bv = bp[ncol + n];
#pragma unroll
  for (int r = 0; r < 8; ++r)
    out[(size_t)(row0 + mbase + r) * OUTD + ncol + n] = acc[r] + bv;
}

// ---------------------------------------------------------------- launcher
extern "C" void kernel_launch(void* const* d_in, const int* in_sizes, int n_in,
                              void* d_out, int out_size, void* d_ws,
                              size_t ws_size, hipStream_t stream) {
  (void)in_sizes; (void)n_in; (void)out_size; (void)ws_size;

  const int*   nfeat    = (const int*)d_in[0];
  const int*   efeat    = (const int*)d_in[1];
  const int*   src      = (const int*)d_in[2];
  const int*   dst      = (const int*)d_in[3];
  const int*   gids     = (const int*)d_in[4];
  const float* atom_emb = (const float*)d_in[5];
  const float* edge_emb = (const float*)d_in[6];  // [5][3][8][256]
  const float* W        = (const float*)d_in[7];  // [5][256][256]
  const float* b        = (const float*)d_in[8];  // [5][256]
  const float* gamma    = (const float*)d_in[9];  // [5][256]
  const float* beta     = (const float*)d_in[10]; // [5][256]
  const float* Wp       = (const float*)d_in[11]; // [256][128]
  const float* bp       = (const float*)d_in[12]; // [128]
  float* out = (float*)d_out;

  // workspace layout (floats)
  float* h        = (float*)d_ws;                   // 12,800,000
  float* neigh    = h + (size_t)N_NODES * EMBED;    // 12,800,000
  float* degs     = neigh + (size_t)N_NODES * EMBED;// 50,000
  float* colsum   = degs + N_NODES;                 // 256
  float* colsumsq = colsum + EMBED;                 // 256
  float* scale    = colsumsq + EMBED;               // 256
  float* shift    = scale + EMBED;                  // 256
  float* gsum     = shift + EMBED;                  // 32,768
  float* gcnt     = gsum + (size_t)N_GRAPHS * EMBED;// 128

  // node embeddings + degrees
  atom_encoder<<<(N_NODES * 64 + 255) / 256, 256, 0, stream>>>(nfeat, atom_emb, h);
  degs_init<<<(N_NODES + 255) / 256, 256, 0, stream>>>(degs);
  degs_acc<<<(N_EDGES + 255) / 256, 256, 0, stream>>>(dst, degs);

  for (int l = 0; l < LAYERS; ++l) {
    zero_f32<<<2048, 256, 0, stream>>>(neigh, N_NODES * EMBED);
    zero_f32<<<2, 256, 0, stream>>>(colsum, 2 * EMBED);  // colsum+colsumsq
    edge_message<<<(N_EDGES * 64 + 255) / 256, 256, 0, stream>>>(
        h, efeat, src, dst,
        edge_emb + (size_t)l * BOND_COLS * BOND_VOCAB * EMBED, neigh);
    gemm_layer<<<N_NODES / 16, 256, 0, stream>>>(
        h, neigh, degs, W + (size_t)l * EMBED * EMBED, b + (size_t)l * EMBED,
        colsum, colsumsq);
    bn_params<<<1, 256, 0, stream>>>(colsum, colsumsq, gamma + (size_t)l * EMBED,
                                     beta + (size_t)l * EMBED, scale, shift);
    bn_relu<<<(N_NODES * EMBED + 255) / 256, 256, 0, stream>>>(neigh, scale,
                                                               shift, h);
  }

  // mean pooling + projection
  zero_f32<<<32, 256, 0, stream>>>(gsum, N_GRAPHS * EMBED + N_GRAPHS);
  pool_sum<<<(N_NODES * 64 + 255) / 256, 256, 0, stream>>>(h, gids, gsum);
  pool_cnt<<<(N_NODES + 255) / 256, 256, 0, stream>>>(gids, gcnt);
  final_gemm<<<N_GRAPHS / 16, 256, 0, stream>>>(gsum, gcnt, Wp, bp, out);
}